// LSTMAutoencoder_80178449482357
// MI455X (gfx1250) — compile-verified
//
#include <hip/hip_runtime.h>

typedef float v2f __attribute__((ext_vector_type(2)));
typedef float v8f __attribute__((ext_vector_type(8)));

#define BB 4096
#define TT 100
#define FF 40
#define HE 64
#define HL 32
#define HD 64

// padded LDS row strides (floats) to avoid bank conflicts
#define GS  260   // gates [16][256]
#define GXS 132   // latent input-gates [16][128]
#define HS  66    // enc/dec hidden [16][64]
#define HLS 34    // lat hidden [16][32]

__device__ __forceinline__ v8f wmma4(v2f a, v2f b, v8f c) {
  // V_WMMA_F32_16X16X4_F32 : D = A(16x4) x B(4x16) + C(16x16)
  return __builtin_amdgcn_wmma_f32_16x16x4_f32(false, a, false, b, (short)0, c,
                                               false, false);
}

__device__ __forceinline__ float sigm(float x) {
  return 1.0f / (1.0f + __expf(-x));
}
__device__ __forceinline__ float tanh_(float x) {
  float e = __expf(2.0f * x);          // robust at both tails
  return 1.0f - 2.0f / (e + 1.0f);
}

__global__ __launch_bounds__(256)
void lstm_ae_kernel(const float* __restrict__ x,
                    const float* __restrict__ eWih, const float* __restrict__ eWhh,
                    const float* __restrict__ eb,
                    const float* __restrict__ lWih, const float* __restrict__ lWhh,
                    const float* __restrict__ lb,
                    const float* __restrict__ dWih, const float* __restrict__ dWhh,
                    const float* __restrict__ db,
                    const float* __restrict__ oW,   const float* __restrict__ ob,
                    float* __restrict__ out) {
  __shared__ float gates[16 * GS];
  __shared__ float gxl[16 * GXS];
  __shared__ float hbuf[16 * HS];   // enc hidden, later dec hidden
  __shared__ float hlat[16 * HLS];  // lat hidden

  const int tid  = threadIdx.x;
  const int lane = tid & 31;
  const int wave = tid >> 5;
  const int lrow = lane & 15;  // M row (A/C) or N col (B/C)
  const int lhi  = lane >> 4;  // half-wave selector
  const int b0   = blockIdx.x * 16;

  // elementwise-phase cell mapping (H=64 layers): 4 cells/thread, same column
  const int r0  = tid >> 6;  // base row 0..3 (rows r0, r0+4, r0+8, r0+12)
  const int j64 = tid & 63;
  // H=32 layer: 2 cells/thread
  const int rl0 = tid >> 5;  // base row 0..7 (rows rl0, rl0+8)
  const int j32 = tid & 31;

  // ---------------- phase-A weights (encoder) into VGPRs ----------------
  v2f wih[10][2];  // enc_Wih [256,40] B-operand slices
  v2f whh[16][2];  // enc_Whh [256,64]
  #pragma unroll
  for (int tt = 0; tt < 2; ++tt) {
    const int gc = wave * 32 + tt * 16;
    #pragma unroll
    for (int kc = 0; kc < 10; ++kc)
      wih[kc][tt] = *(const v2f*)&eWih[(gc + lrow) * FF + kc * 4 + 2 * lhi];
    #pragma unroll
    for (int kc = 0; kc < 16; ++kc)
      whh[kc][tt] = *(const v2f*)&eWhh[(gc + lrow) * HE + kc * 4 + 2 * lhi];
  }
  const float ebi = eb[j64], ebf = eb[64 + j64], ebg = eb[128 + j64], ebo = eb[192 + j64];

  for (int i = tid; i < 16 * HS; i += 256) hbuf[i] = 0.0f;
  float cE[4] = {0.f, 0.f, 0.f, 0.f};
  __syncthreads();

  const float* xblk = x + (size_t)b0 * TT * FF;

  // ============================ encoder ============================
  #pragma unroll 1
  for (int t = 0; t < TT; ++t) {
    v8f acc0 = {}, acc1 = {};
    const float* xt = xblk + (size_t)lrow * TT * FF + t * FF + 2 * lhi;
    #pragma unroll
    for (int kc = 0; kc < 10; ++kc) {        // x_t @ Wih^T  (K=40)
      v2f a = *(const v2f*)&xt[kc * 4];
      acc0 = wmma4(a, wih[kc][0], acc0);
      acc1 = wmma4(a, wih[kc][1], acc1);
    }
    #pragma unroll
    for (int kc = 0; kc < 16; ++kc) {        // h @ Whh^T  (K=64)
      v2f a = *(const v2f*)&hbuf[lrow * HS + kc * 4 + 2 * lhi];
      acc0 = wmma4(a, whh[kc][0], acc0);
      acc1 = wmma4(a, whh[kc][1], acc1);
    }
    const int gc0 = wave * 32;
    #pragma unroll
    for (int jj = 0; jj < 8; ++jj) {
      gates[(jj + 8 * lhi) * GS + gc0 + lrow]      = acc0[jj];
      gates[(jj + 8 * lhi) * GS + gc0 + 16 + lrow] = acc1[jj];
    }
    __syncthreads();
    #pragma unroll
    for (int i = 0; i < 4; ++i) {
      const int r = r0 + 4 * i;
      float gi = gates[r * GS + j64] + ebi;
      float gf = gates[r * GS + 64 + j64] + ebf;
      float gg = gates[r * GS + 128 + j64] + ebg;
      float go = gates[r * GS + 192 + j64] + ebo;
      float c  = sigm(gf) * cE[i] + sigm(gi) * tanh_(gg);
      cE[i] = c;
      hbuf[r * HS + j64] = sigm(go) * tanh_(c);
    }
    __syncthreads();
  }

  // ---- latent input gates: constant over t (input is broadcast last h) ----
  {
    const int gc = wave * 16;  // 8 waves cover 128 gate cols
    v8f acc = {};
    #pragma unroll
    for (int kc = 0; kc < 16; ++kc) {        // last_h @ lat_Wih^T (K=64)
      v2f a = *(const v2f*)&hbuf[lrow * HS + kc * 4 + 2 * lhi];
      v2f b = *(const v2f*)&lWih[(gc + lrow) * HE + kc * 4 + 2 * lhi];
      acc = wmma4(a, b, acc);
    }
    const float bias = lb[gc + lrow];
    #pragma unroll
    for (int jj = 0; jj < 8; ++jj)
      gxl[(jj + 8 * lhi) * GXS + gc + lrow] = acc[jj] + bias;
  }
  __syncthreads();  // everyone done reading hbuf (enc last h) & writing gxl

  // ---------------- phase-B weights into VGPRs ----------------
  v2f wlh[8];       // lat_Whh [128,32]
  v2f dwi[8][2];    // dec_Wih [256,32]
  v2f dwh[16][2];   // dec_Whh [256,64]
  v2f wo[16];       // out_W   [40,64] (waves 0..2 only, cols masked)
  {
    const int gcl = wave * 16;
    #pragma unroll
    for (int kc = 0; kc < 8; ++kc)
      wlh[kc] = *(const v2f*)&lWhh[(gcl + lrow) * HL + kc * 4 + 2 * lhi];
    #pragma unroll
    for (int tt = 0; tt < 2; ++tt) {
      const int gc = wave * 32 + tt * 16;
      #pragma unroll
      for (int kc = 0; kc < 8; ++kc)
        dwi[kc][tt] = *(const v2f*)&dWih[(gc + lrow) * HL + kc * 4 + 2 * lhi];
      #pragma unroll
      for (int kc = 0; kc < 16; ++kc)
        dwh[kc][tt] = *(const v2f*)&dWhh[(gc + lrow) * HD + kc * 4 + 2 * lhi];
    }
    const int f = wave * 16 + lrow;
    #pragma unroll
    for (int kc = 0; kc < 16; ++kc) {
      v2f z = {};
      if (wave < 3 && f < FF)
        z = *(const v2f*)&oW[f * HD + kc * 4 + 2 * lhi];
      wo[kc] = z;
    }
  }
  const float dbi = db[j64], dbf = db[64 + j64], dbg = db[128 + j64], dbo = db[192 + j64];
  const float obias = (wave < 3 && (wave * 16 + lrow) < FF) ? ob[wave * 16 + lrow] : 0.0f;

  for (int i = tid; i < 16 * HS; i += 256) hbuf[i] = 0.0f;   // dec hidden
  for (int i = tid; i < 16 * HLS; i += 256) hlat[i] = 0.0f;  // lat hidden
  float cL[2] = {0.f, 0.f};
  float cD[4] = {0.f, 0.f, 0.f, 0.f};
  __syncthreads();

  float* outblk = out + (size_t)b0 * TT * FF;

  // ================= fused latent -> decoder -> projection =================
  #pragma unroll 1
  for (int t = 0; t < TT; ++t) {
    // latent recurrent gates (K=32), 1 N-tile per wave
    {
      const int gc = wave * 16;
      v8f acc = {};
      #pragma unroll
      for (int kc = 0; kc < 8; ++kc) {
        v2f a = *(const v2f*)&hlat[lrow * HLS + kc * 4 + 2 * lhi];
        acc = wmma4(a, wlh[kc], acc);
      }
      #pragma unroll
      for (int jj = 0; jj < 8; ++jj)
        gates[(jj + 8 * lhi) * GS + gc + lrow] = acc[jj];
    }
    __syncthreads();
    #pragma unroll
    for (int i = 0; i < 2; ++i) {            // latent elementwise (H=32)
      const int r = rl0 + 8 * i;
      float gi = gates[r * GS + j32]      + gxl[r * GXS + j32];
      float gf = gates[r * GS + 32 + j32] + gxl[r * GXS + 32 + j32];
      float gg = gates[r * GS + 64 + j32] + gxl[r * GXS + 64 + j32];
      float go = gates[r * GS + 96 + j32] + gxl[r * GXS + 96 + j32];
      float c  = sigm(gf) * cL[i] + sigm(gi) * tanh_(gg);
      cL[i] = c;
      hlat[r * HLS + j32] = sigm(go) * tanh_(c);
    }
    __syncthreads();
    // decoder gates: h_lat @ dec_Wih^T (K=32) + h_dec @ dec_Whh^T (K=64)
    {
      v8f acc0 = {}, acc1 = {};
      #pragma unroll
      for (int kc = 0; kc < 8; ++kc) {
        v2f a = *(const v2f*)&hlat[lrow * HLS + kc * 4 + 2 * lhi];
        acc0 = wmma4(a, dwi[kc][0], acc0);
        acc1 = wmma4(a, dwi[kc][1], acc1);
      }
      #pragma unroll
      for (int kc = 0; kc < 16; ++kc) {
        v2f a = *(const v2f*)&hbuf[lrow * HS + kc * 4 + 2 * lhi];
        acc0 = wmma4(a, dwh[kc][0], acc0);
        acc1 = wmma4(a, dwh[kc][1], acc1);
      }
      const int gc0 = wave * 32;
      #pragma unroll
      for (int jj = 0; jj < 8; ++jj) {
        gates[(jj + 8 * lhi) * GS + gc0 + lrow]      = acc0[jj];
        gates[(jj + 8 * lhi) * GS + gc0 + 16 + lrow] = acc1[jj];
      }
    }
    __syncthreads();
    #pragma unroll
    for (int i = 0; i < 4; ++i) {            // decoder elementwise (H=64)
      const int r = r0 + 4 * i;
      float gi = gates[r * GS + j64] + dbi;
      float gf = gates[r * GS + 64 + j64] + dbf;
      float gg = gates[r * GS + 128 + j64] + dbg;
      float go = gates[r * GS + 192 + j64] + dbo;
      float c  = sigm(gf) * cD[i] + sigm(gi) * tanh_(gg);
      cD[i] = c;
      hbuf[r * HS + j64] = sigm(go) * tanh_(c);
    }
    __syncthreads();
    // output projection: out[b,t,:] = h_dec @ out_W^T + out_b  (waves 0..2)
    if (wave < 3) {
      v8f acc = {};
      #pragma unroll
      for (int kc = 0; kc < 16; ++kc) {
        v2f a = *(const v2f*)&hbuf[lrow * HS + kc * 4 + 2 * lhi];
        acc = wmma4(a, wo[kc], acc);
      }
      const int f = wave * 16 + lrow;
      if (f < FF) {
        #pragma unroll
        for (int jj = 0; jj < 8; ++jj) {
          const int m = jj + 8 * lhi;
          outblk[(size_t)m * TT * FF + t * FF + f] = acc[jj] + obias;
        }
      }
    }
    // no barrier needed: next-iter writes (gates) are disjoint from hbuf reads,
    // and all gates readers sit behind the next barrier.
  }
}

extern "C" void kernel_launch(void* const* d_in, const int* in_sizes, int n_in,
                              void* d_out, int out_size, void* d_ws, size_t ws_size,
                              hipStream_t stream) {
  (void)in_sizes; (void)n_in; (void)out_size; (void)d_ws; (void)ws_size;
  const float* x    = (const float*)d_in[0];
  const float* eWih = (const float*)d_in[1];
  const float* eWhh = (const float*)d_in[2];
  const float* eb   = (const float*)d_in[3];
  const float* lWih = (const float*)d_in[4];
  const float* lWhh = (const float*)d_in[5];
  const float* lb   = (const float*)d_in[6];
  const float* dWih = (const float*)d_in[7];
  const float* dWhh = (const float*)d_in[8];
  const float* db   = (const float*)d_in[9];
  const float* oW   = (const float*)d_in[10];
  const float* ob   = (const float*)d_in[11];
  float* out = (float*)d_out;

  dim3 grid(BB / 16), block(256);
  hipLaunchKernelGGL(lstm_ae_kernel, grid, block, 0, stream,
                     x, eWih, eWhh, eb, lWih, lWhh, lb, dWih, dWhh, db, oW, ob, out);
}